// Style_dubber_model_loss_15_GRID_30623116820871
// MI455X (gfx1250) — compile-verified
//
#include <hip/hip_runtime.h>

#define B_    32
#define T_    512
#define M_    2048
#define L_    1024
#define NMEL_ 80
#define DSPK_ 256
#define NB1   1024   // mel partial blocks
#define BPB   64     // attn blocks per batch

typedef float f4  __attribute__((ext_vector_type(4)));
typedef float v2f __attribute__((ext_vector_type(2)));
typedef float v8f __attribute__((ext_vector_type(8)));

// Wave32 sum via V_WMMA_F32_16X16X4_F32 with B == all-ones.
// A layout (16x4 f32): lane m(0-15): VGPR0=A[m][0], VGPR1=A[m][1];
// lane m+16: VGPR0=A[m][2], VGPR1=A[m][3]. With a={v,0}, B=1:
// D[m][n] = v_m + v_{m+16}. Per-lane sum of the 8 D VGPRs covers rows
// 0-7 (lanes 0-15) / 8-15 (lanes 16-31); one shfl_xor(16) completes it.
__device__ __forceinline__ float wave_reduce_sum(float v) {
  v2f a; a[0] = v;    a[1] = 0.0f;
  v2f b; b[0] = 1.0f; b[1] = 1.0f;
  v8f c = {0.f, 0.f, 0.f, 0.f, 0.f, 0.f, 0.f, 0.f};
  c = __builtin_amdgcn_wmma_f32_16x16x4_f32(false, a, false, b, (short)0, c,
                                            false, false);
  float t = c[0] + c[1] + c[2] + c[3] + c[4] + c[5] + c[6] + c[7];
  t += __shfl_xor(t, 16, 32);
  return t;
}

// 256-thread block sum (8 waves). Safe to call repeatedly with same lds.
__device__ __forceinline__ float block_reduce256(float v, float* lds) {
  float w = wave_reduce_sum(v);
  __syncthreads();
  if ((threadIdx.x & 31) == 0) lds[threadIdx.x >> 5] = w;
  __syncthreads();
  float r = 0.f;
#pragma unroll
  for (int i = 0; i < 8; ++i) r += lds[i];
  return r;
}

// ---------------- Kernel 1: masked mel MAE partial sums ----------------
__global__ __launch_bounds__(256) void mel_mae_kernel(
    const f4* __restrict__ tgt, const f4* __restrict__ pred,
    const f4* __restrict__ post, const int* __restrict__ mel_lens,
    float* __restrict__ part1, float* __restrict__ part2) {
  __shared__ float lds[8];
  __shared__ int slen[B_];
  if (threadIdx.x < B_) slen[threadIdx.x] = mel_lens[threadIdx.x];
  __syncthreads();
  const int n4 = B_ * M_ * NMEL_ / 4;  // NMEL % 4 == 0 -> (b,m) uniform per f4
  float s1 = 0.f, s2 = 0.f;
  for (int i = blockIdx.x * blockDim.x + threadIdx.x; i < n4;
       i += gridDim.x * blockDim.x) {
    int e = i << 2;
    int b = e / (M_ * NMEL_);
    int m = (e / NMEL_) % M_;
    if (m < slen[b]) {  // skip padded rows: saves bandwidth, contributes 0
      f4 t = __builtin_nontemporal_load(&tgt[i]);
      f4 p = __builtin_nontemporal_load(&pred[i]);
      f4 q = __builtin_nontemporal_load(&post[i]);
      s1 += fabsf(p[0]-t[0]) + fabsf(p[1]-t[1]) + fabsf(p[2]-t[2]) + fabsf(p[3]-t[3]);
      s2 += fabsf(q[0]-t[0]) + fabsf(q[1]-t[1]) + fabsf(q[2]-t[2]) + fabsf(q[3]-t[3]);
    }
  }
  float r1 = block_reduce256(s1, lds);
  float r2 = block_reduce256(s2, lds);
  if (threadIdx.x == 0) { part1[blockIdx.x] = r1; part2[blockIdx.x] = r2; }
}

// ---------- Kernel 2: diagonal-attention num/den partial sums ----------
// attn_scores is [B,T,L]; reference's attn[b,l,t] == attn_scores[b,t,l].
__global__ __launch_bounds__(256) void attn_kernel(
    const f4* __restrict__ attn, const int* __restrict__ xl,
    const int* __restrict__ ll, float* __restrict__ pnum,
    float* __restrict__ pden) {
  __shared__ float lds[8];
  const int b   = blockIdx.x / BPB;
  const int sub = blockIdx.x % BPB;
  const int xlen = xl[b];
  const int llen = ll[b];
  const float ks = (float)llen / (float)xlen;
  const float da = (float)llen / 8.0f;
  const int Jb    = T_ * L_ / 4;   // 131072 float4 per batch
  const int chunk = Jb / BPB;      // 2048 per block
  const f4* base = attn + (size_t)b * Jb;
  float num = 0.f, den = 0.f;
#pragma unroll
  for (int it = 0; it < chunk / 256; ++it) {
    int j = sub * chunk + it * 256 + (int)threadIdx.x;
    int t = j >> 8;                 // L/4 == 256 -> t uniform across the wave
    if (t < xlen) {                 // uniform branch: whole padded rows skipped
      int l0 = (j & 255) << 2;
      if (l0 < llen) {
        f4 a = __builtin_nontemporal_load(&base[j]);
        float ft = (float)t;
        float y1 = floorf(ks * ft + da);
        float y2 = floorf(fmaxf(ks * ft - da, 0.f));
#pragma unroll
        for (int k = 0; k < 4; ++k) {
          int l = l0 + k;
          if (l < llen) {
            float av = a[k];
            den += av;
            float fl = (float)l;
            if (fl >= y2 && fl < y1) num += av;
          }
        }
      }
    }
  }
  float rn = block_reduce256(num, lds);
  float rd = block_reduce256(den, lds);
  if (threadIdx.x == 0) { pnum[blockIdx.x] = rn; pden[blockIdx.x] = rd; }
}

// --------- Kernel 3: finalize (duration, speaker, combine) -------------
__global__ __launch_bounds__(256) void finalize_kernel(
    const float* __restrict__ part1, const float* __restrict__ part2,
    const float* __restrict__ pnum, const float* __restrict__ pden,
    const int* __restrict__ mel_lens, const int* __restrict__ xl,
    const float* __restrict__ ldp, const int* __restrict__ dur,
    const float* __restrict__ spk_p, const float* __restrict__ spk_e,
    float* __restrict__ out) {
  __shared__ float lds[8];
  __shared__ float sA[B_][8];
  __shared__ float sB[B_][8];
  __shared__ float sC[B_][8];
  __shared__ float sScalar[B_];
  const int tid = threadIdx.x;

  // mel partial reduce
  float v1 = 0.f, v2 = 0.f;
  for (int i = tid; i < NB1; i += 256) { v1 += part1[i]; v2 += part2[i]; }
  float mel_sum  = block_reduce256(v1, lds);
  float post_sum = block_reduce256(v2, lds);
  float cnt_mel  = block_reduce256(tid < B_ ? (float)mel_lens[tid] : 0.f, lds);

  // masked log-domain duration MSE
  float dv = 0.f;
  for (int i = tid; i < B_ * T_; i += 256) {
    int b = i / T_, t = i % T_;
    if (t < xl[b]) {
      float d = ldp[i] - logf((float)dur[i] + 1.0f);
      dv += d * d;
    }
  }
  float dur_num = block_reduce256(dv, lds);
  float cnt_dur = block_reduce256(tid < B_ ? (float)xl[tid] : 0.f, lds);

  // speaker cosine: 8 threads per batch over 256 dims
  {
    int b = tid >> 3, j0 = tid & 7;
    float dot = 0.f, n1 = 0.f, n2 = 0.f;
    for (int k = j0; k < DSPK_; k += 8) {
      float a = spk_p[b * DSPK_ + k];
      float e = spk_e[b * DSPK_ + k];
      dot += a * e; n1 += a * a; n2 += e * e;
    }
    sA[b][j0] = dot; sB[b][j0] = n1; sC[b][j0] = n2;
  }
  __syncthreads();
  if (tid < B_) {
    float dot = 0.f, n1 = 0.f, n2 = 0.f;
#pragma unroll
    for (int j = 0; j < 8; ++j) { dot += sA[tid][j]; n1 += sB[tid][j]; n2 += sC[tid][j]; }
    const float eps = 1e-6f;
    float cosv = dot / (fmaxf(sqrtf(n1), eps) * fmaxf(sqrtf(n2), eps));
    sScalar[tid] = 1.0f - cosv;
  }
  __syncthreads();
  float spk_sum = block_reduce256(tid < B_ ? sScalar[tid] : 0.f, lds);

  // attention focus: reduce BPB partials per batch, then mean over b
  __syncthreads();
  {
    int b = tid >> 3, j0 = tid & 7;
    float n = 0.f, d = 0.f;
#pragma unroll
    for (int s = j0 * 8; s < j0 * 8 + 8; ++s) {
      n += pnum[b * BPB + s];
      d += pden[b * BPB + s];
    }
    sA[b][j0] = n; sB[b][j0] = d;
  }
  __syncthreads();
  if (tid < B_) {
    float n = 0.f, d = 0.f;
#pragma unroll
    for (int j = 0; j < 8; ++j) { n += sA[tid][j]; d += sB[tid][j]; }
    sScalar[tid] = n / d;
  }
  __syncthreads();
  float focus_sum = block_reduce256(tid < B_ ? sScalar[tid] : 0.f, lds);

  if (tid == 0) {
    float cntm      = cnt_mel * (float)NMEL_;
    float mel_loss  = mel_sum  / cntm;
    float post_loss = post_sum / cntm;
    float dur_loss  = dur_num  / cnt_dur;
    float spk_loss  = spk_sum  / (float)B_;
    float dia_loss  = 0.1f * -logf(focus_sum / (float)B_);
    out[1] = mel_loss;
    out[2] = post_loss;
    out[3] = spk_loss;
    out[4] = dur_loss;
    out[5] = dia_loss;
    out[0] = mel_loss + post_loss + dur_loss + spk_loss + dia_loss;
  }
}

extern "C" void kernel_launch(void* const* d_in, const int* in_sizes, int n_in,
                              void* d_out, int out_size, void* d_ws, size_t ws_size,
                              hipStream_t stream) {
  const float* mel_targets = (const float*)d_in[2];
  const int*   mel_lens    = (const int*)  d_in[5];
  const int*   dur_t       = (const int*)  d_in[6];
  const float* mel_pred    = (const float*)d_in[8];
  const float* post_pred   = (const float*)d_in[9];
  const float* ldp         = (const float*)d_in[10];
  const float* spk_p       = (const float*)d_in[14];
  const float* spk_e       = (const float*)d_in[15];
  const float* attn        = (const float*)d_in[16];
  const int*   xl          = (const int*)  d_in[17];
  const int*   ll          = (const int*)  d_in[18];

  float* ws    = (float*)d_ws;
  float* part1 = ws;                       // NB1
  float* part2 = ws + NB1;                 // NB1
  float* pnum  = ws + 2 * NB1;             // B_*BPB
  float* pden  = ws + 2 * NB1 + B_ * BPB;  // B_*BPB

  mel_mae_kernel<<<NB1, 256, 0, stream>>>(
      (const f4*)mel_targets, (const f4*)mel_pred, (const f4*)post_pred,
      mel_lens, part1, part2);
  attn_kernel<<<B_ * BPB, 256, 0, stream>>>(
      (const f4*)attn, xl, ll, pnum, pden);
  finalize_kernel<<<1, 256, 0, stream>>>(
      part1, part2, pnum, pden, mel_lens, xl, ldp, dur_t, spk_p, spk_e,
      (float*)d_out);
}